// ExponentialSmoothingAttention_84164179133244
// MI455X (gfx1250) — compile-verified
//
#include <hip/hip_runtime.h>

typedef __attribute__((ext_vector_type(2))) float v2f;
typedef __attribute__((ext_vector_type(8))) float v8f;

#define LSEQ   8192
#define DMODEL 1024
#define ROWS_PER_WAVE 512
#define STORE_TILES   (ROWS_PER_WAVE / 16)        // 32
#define WARM_TILES    10                          // 160-row lookahead: (1-a)^160 ~ 1e-67
#define TOT_TILES     (STORE_TILES + WARM_TILES)  // 42

struct Aregs { v2f A0, A1, A2, A3, Ac; };

// Vext[m]: m==0 -> v0 ; 1..L -> V[m-1] ; m>L -> 0   (used only for the two static edge tiles)
__device__ __forceinline__ float x_load_guard(const float* __restrict__ V,
                                              const float* __restrict__ v0,
                                              int b, int m, int c) {
    if (m > LSEQ) return 0.0f;
    if (m == 0)   return v0[c];
    return V[((size_t)b * LSEQ + (size_t)(m - 1)) * DMODEL + c];
}

// 16x16 tile: Y = T(triangular decay) * X + rank-1 carry; returns acc, updates carry (B layout)
__device__ __forceinline__ v8f tile_mma(const Aregs& A, const v2f B[4], v2f& bc) {
    v8f acc = {};
    acc = __builtin_amdgcn_wmma_f32_16x16x4_f32(false, A.A0, false, B[0], (short)0, acc, false, false);
    acc = __builtin_amdgcn_wmma_f32_16x16x4_f32(false, A.A1, false, B[1], (short)0, acc, false, false);
    acc = __builtin_amdgcn_wmma_f32_16x16x4_f32(false, A.A2, false, B[2], (short)0, acc, false, false);
    acc = __builtin_amdgcn_wmma_f32_16x16x4_f32(false, A.A3, false, B[3], (short)0, acc, false, false);
    acc = __builtin_amdgcn_wmma_f32_16x16x4_f32(false, A.Ac, false, bc,   (short)0, acc, false, false);
    // next tile's carry S_s = Y[row 0] = acc VGPR0 (lanes 16-31 garbage hits zero A.Ac coefs)
    bc.x = acc[0];
    bc.y = 0.0f;
    return acc;
}

// branch-free tile load: p points at Vext row (s + koff), one column; row stride = DMODEL
__device__ __forceinline__ void load_tile_fast(const float* __restrict__ p, v2f B[4]) {
#pragma unroll
    for (int j = 0; j < 4; ++j) {
        B[j].x = p[(size_t)(4 * j) * DMODEL];
        B[j].y = p[(size_t)(4 * j + 1) * DMODEL];
    }
}

__device__ __forceinline__ void load_tile_guard(const float* __restrict__ V,
                                                const float* __restrict__ v0,
                                                int b, int s, int koff, int c, v2f B[4]) {
#pragma unroll
    for (int j = 0; j < 4; ++j) {
        B[j].x = x_load_guard(V, v0, b, s + 4 * j + koff, c);
        B[j].y = x_load_guard(V, v0, b, s + 4 * j + 1 + koff, c);
    }
}

__device__ __forceinline__ void store_tile(float* __restrict__ out, int b, int s, int c,
                                           int lane, const v8f& acc) {
    const int rbase = s + ((lane < 16) ? 0 : 8);
    const size_t o  = ((size_t)b * LSEQ + (size_t)rbase) * DMODEL + (size_t)c;
#pragma unroll
    for (int r = 0; r < 8; ++r)
        out[o + (size_t)r * DMODEL] = acc[r];
}

__global__ void __launch_bounds__(256)
esa_suffix_scan_wmma(const float* __restrict__ V,
                     const float* __restrict__ alpha,
                     const float* __restrict__ v0,
                     float* __restrict__ out) {
    const int lane = threadIdx.x & 31;
    const int wid  = threadIdx.x >> 5;
    const int gw   = blockIdx.x * 8 + wid;        // 4096 waves
    const int b    = gw >> 10;                    // 4 batches
    const int rem  = gw & 1023;
    const int c0   = (rem >> 4) * 16;             // 64 column groups (uniform head per group)
    const int r0   = (rem & 15) * ROWS_PER_WAVE;  // 16 row chunks

    const int   head = c0 >> 6;
    const float a    = 1.0f / (1.0f + expf(-alpha[head]));
    const float oma  = 1.0f - a;

    const int M    = lane & 15;
    const int koff = (lane < 16) ? 0 : 2;
    const int c    = c0 + M;

    // A: triangular decay T[M, m] = a*(1-a)^(m-M) for m>=M (m = 4j + K)
    Aregs A;
    {
        v2f* Aj[4] = { &A.A0, &A.A1, &A.A2, &A.A3 };
#pragma unroll
        for (int j = 0; j < 4; ++j) {
            const int m0 = 4 * j + koff;
            const int m1 = m0 + 1;
            Aj[j]->x = (m0 >= M) ? a * __powf(oma, (float)(m0 - M)) : 0.0f;
            Aj[j]->y = (m1 >= M) ? a * __powf(oma, (float)(m1 - M)) : 0.0f;
        }
        A.Ac.x = (lane < 16) ? __powf(oma, (float)(16 - M)) : 0.0f;
        A.Ac.y = 0.0f;
    }

    v2f bc = {0.0f, 0.0f};

    // clip warm-up: tiles entirely past m=L contribute exactly 0
    int i_hi = min(TOT_TILES - 1, (LSEQ - r0) >> 4);

    // static edge tile at s == LSEQ (top row-chunk only): needs m>L guard; never stored
    if (r0 + 16 * i_hi + 15 > LSEQ) {
        const int s = r0 + 16 * i_hi;
        v2f B[4];
        load_tile_guard(V, v0, b, s, koff, c, B);
        v8f acc = tile_mma(A, B, bc);
        if (i_hi < STORE_TILES) store_tile(out, b, s, c, lane, acc);
        --i_hi;
    }

    const int i_lo = (r0 == 0) ? 1 : 0;   // s==0 tile (contains v0) handled separately

    // hot loop: branch-free clauses of 8 loads, double-buffered against the WMMA chain
    if (i_hi >= i_lo) {
        v2f Bcur[4], Bnext[4];
        {
            const int s0 = r0 + 16 * i_hi;
            load_tile_fast(V + ((size_t)b * LSEQ + (size_t)(s0 - 1 + koff)) * DMODEL + c, Bcur);
        }
        for (int i = i_hi; i >= i_lo; --i) {
            if (i > i_lo) {
                const int sn = r0 + 16 * (i - 1);
                load_tile_fast(V + ((size_t)b * LSEQ + (size_t)(sn - 1 + koff)) * DMODEL + c, Bnext);
            }
            v8f acc = tile_mma(A, Bcur, bc);
            if (i < STORE_TILES)
                store_tile(out, b, r0 + 16 * i, c, lane, acc);
#pragma unroll
            for (int j = 0; j < 4; ++j) Bcur[j] = Bnext[j];
        }
    }

    // static edge tile s == 0 (row chunk 0 only): row m=0 is v0
    if (r0 == 0) {
        v2f B[4];
        load_tile_guard(V, v0, b, 0, koff, c, B);
        v8f acc = tile_mma(A, B, bc);
        store_tile(out, b, 0, c, lane, acc);
    }
}

extern "C" void kernel_launch(void* const* d_in, const int* in_sizes, int n_in,
                              void* d_out, int out_size, void* d_ws, size_t ws_size,
                              hipStream_t stream) {
    (void)in_sizes; (void)n_in; (void)d_ws; (void)ws_size; (void)out_size;
    const float* V     = (const float*)d_in[0];   // (4, 8192, 1024) f32
    const float* alpha = (const float*)d_in[1];   // (16,) f32
    const float* v0    = (const float*)d_in[2];   // (16, 64) f32
    float* out = (float*)d_out;                   // (4, 8192, 1024) f32

    // 4096 waves = 512 blocks * 8 waves; each wave: 16 cols x 512 rows + 160-row warmup
    esa_suffix_scan_wmma<<<512, 256, 0, stream>>>(V, alpha, v0, out);
}